// RGCN_27994596836125
// MI455X (gfx1250) — compile-verified
//
#include <hip/hip_runtime.h>
#include <math.h>

typedef __attribute__((ext_vector_type(2))) float v2f;
typedef __attribute__((ext_vector_type(8))) float v8f;

#define NN 50000
#define EE 800000
#define RR 8

// ---------------------------------------------------------------------------
// CDNA5 async global->LDS staging (ASYNCcnt path), with sync fallback
// ---------------------------------------------------------------------------
typedef int b128vec __attribute__((vector_size(16)));
#define AS1 __attribute__((address_space(1)))
#define AS3 __attribute__((address_space(3)))

#if __has_builtin(__builtin_amdgcn_global_load_async_to_lds_b128)
#define HAVE_ASYNC_LDS 1
#define ASYNC_COPY_B128(g, l)                                                  \
    __builtin_amdgcn_global_load_async_to_lds_b128(                            \
        (AS1 b128vec*)(g), (AS3 b128vec*)(l), 0, 0)
#else
#define HAVE_ASYNC_LDS 0
#define ASYNC_COPY_B128(g, l) *(float4*)(l) = *(const float4*)(g)
#endif

#if HAVE_ASYNC_LDS
#if __has_builtin(__builtin_amdgcn_s_wait_asynccnt)
#define WAIT_ASYNC() __builtin_amdgcn_s_wait_asynccnt(0)
#else
#define WAIT_ASYNC() asm volatile("s_wait_asynccnt 0x0" ::: "memory")
#endif
#else
#define WAIT_ASYNC()
#endif

// ---------------------------------------------------------------------------
// utility kernels
// ---------------------------------------------------------------------------
__global__ void zero_f32(float* __restrict__ p, int n) {
    int i = blockIdx.x * blockDim.x + threadIdx.x;
    int stride = gridDim.x * blockDim.x;
    for (; i < n; i += stride) p[i] = 0.0f;
}

__global__ void relu_inplace(float* __restrict__ p, int n) {
    int i = blockIdx.x * blockDim.x + threadIdx.x;
    int stride = gridDim.x * blockDim.x;
    for (; i < n; i += stride) p[i] = fmaxf(p[i], 0.0f);
}

__global__ void count_edges(const int* __restrict__ ei, const int* __restrict__ et,
                            float* __restrict__ cnt, int E, int N) {
    int e = blockIdx.x * blockDim.x + threadIdx.x;
    if (e >= E) return;
    int dst = ei[E + e];
    int r = et[e];
    atomicAdd(&cnt[(size_t)r * N + dst], 1.0f);
}

__global__ void finalize_inv(float* __restrict__ cnt, int n) {
    int i = blockIdx.x * blockDim.x + threadIdx.x;
    if (i >= n) return;
    cnt[i] = 1.0f / fmaxf(cnt[i], 1.0f);
}

// one wave per edge; lanes stripe the F features; only edges of relation `rel`
__global__ void scatter_rel(const float* __restrict__ x, const int* __restrict__ ei,
                            const int* __restrict__ et, float* __restrict__ buf,
                            int rel, int F, int E) {
    int gid = blockIdx.x * blockDim.x + threadIdx.x;
    int wid = gid >> 5;
    int lane = threadIdx.x & 31;
    if (wid >= E) return;
    if (et[wid] != rel) return;
    int src = ei[wid];
    int dst = ei[E + wid];
    const float* xs = x + (size_t)src * F;
    float* bd = buf + (size_t)dst * F;
    for (int f = lane; f < F; f += 32) atomicAdd(&bd[f], xs[f]);
}

// ---------------------------------------------------------------------------
// Pre-pack W[K,O] (count matrices) into paired-row layout:
//   Wp[p*O + c][2] = { W[2p][c], W[2p+1][c] }
// so a 16B async copy from Wp lands two (row,row+1) column pairs contiguously
// in LDS, making each WMMA B fragment a single contiguous ds_load_b64.
// ---------------------------------------------------------------------------
__global__ void pack_pairs(const float* __restrict__ src, float* __restrict__ dst,
                           int K, int O, int count) {
    size_t per = (size_t)(K / 2) * O;
    size_t total = per * count;
    size_t i = (size_t)blockIdx.x * blockDim.x + threadIdx.x;
    size_t stride = (size_t)gridDim.x * blockDim.x;
    for (; i < total; i += stride) {
        size_t m = i / per;
        size_t rem = i - m * per;
        size_t p = rem / O, c = rem - p * O;
        const float* s = src + m * (size_t)K * O;
        float* d = dst + m * (size_t)K * O;
        float2 v = make_float2(s[(2 * p) * (size_t)O + c],
                               s[(2 * p + 1) * (size_t)O + c]);
        *(float2*)(d + ((p * (size_t)O) + c) * 2) = v;
    }
}

// ---------------------------------------------------------------------------
// fp32 WMMA GEMM:  out[N,O] (+)= (A[N,K] * rowscale) @ W[K,O]  (+ bias)
//
// Block = 8 waves sharing one 64-wide O strip; each wave owns a 16-row M tile.
// K stepped in KC=32 chunks, double-buffered in LDS: async global->LDS copies
// for chunk c+1 are issued before computing chunk c (V_WMMA_F32_16X16X4_F32
// fed from conflict-free ds_load_b64 fragments), then s_wait_asynccnt+barrier.
//   A stride 36 floats -> b64 frag banks all distinct across 32 lanes.
//   B paired-row stride 160 floats (=32 mod 64) -> lane halves 32 banks apart.
// ---------------------------------------------------------------------------
#define KC 32
#define SA 36    // A LDS row stride (floats)
#define PS 160   // B LDS pair-row stride (floats): 128 data + 32 pad

__global__ __launch_bounds__(256) void rgcn_gemm_f32(
    const float* __restrict__ Ag, const float* __restrict__ inv,
    const float* __restrict__ Wp, const float* __restrict__ bias,
    float* __restrict__ out, int ntiles, int K, int O, int accumulate)
{
    __shared__ float Bs[2][(KC / 2) * PS];   // 2 x 10 KB
    __shared__ float As[2][8 * 16 * SA];     // 2 x 18 KB

    int tid = threadIdx.x;
    int wave = tid >> 5;
    int lane = tid & 31;
    int mtile = blockIdx.x * 8 + wave;
    bool active = (mtile < ntiles);          // wave-uniform

    int m0 = mtile << 4;
    int o0 = blockIdx.y << 6;                // 64 output cols per block
    int lm = lane & 15;
    int hi = lane >> 4;                      // 0 or 1
    int kb = hi << 1;                        // K sub-offset 0 or 2
    int row = m0 + lm;
    int O2 = O << 1;                         // packed-W pair-row stride

    float s = 1.0f;
    if (active && inv) s = inv[row];

    auto stageB = [&](int kc, int bi) {
        for (int j = 0; j < 2; ++j) {
            int idx = tid + (j << 8);              // 0..511 b128s
            int pr = idx >> 5;                     // pair row 0..15
            int cq = (idx & 31) << 2;              // 0..124 floats
            const float* g = Wp + (size_t)((kc >> 1) + pr) * O2 + (o0 << 1) + cq;
            float* l = &Bs[bi][pr * PS + cq];
            ASYNC_COPY_B128(g, l);
        }
    };
    auto stageA = [&](int kc, int bi) {
        if (!active) return;
        for (int j = 0; j < 4; ++j) {
            int idx = lane + (j << 5);             // 0..127 b128s
            int arow = idx >> 3;                   // 0..15
            int ac = (idx & 7) << 2;               // 0..28 floats
            const float* g = Ag + (size_t)(m0 + arow) * K + kc + ac;
            float* l = &As[bi][wave * 16 * SA + arow * SA + ac];
            ASYNC_COPY_B128(g, l);
        }
    };

    v8f acc[4];
    if (active) {
        if (accumulate) {
            for (int t = 0; t < 4; ++t)
                for (int i = 0; i < 8; ++i)
                    acc[t][i] = out[(size_t)(m0 + i + 8 * hi) * O + o0 + t * 16 + lm];
        } else if (bias) {
            for (int t = 0; t < 4; ++t) {
                float bv = bias[o0 + t * 16 + lm];
                for (int i = 0; i < 8; ++i) acc[t][i] = bv;
            }
        } else {
            for (int t = 0; t < 4; ++t)
                for (int i = 0; i < 8; ++i) acc[t][i] = 0.0f;
        }
    }

    // prologue: stage chunk 0
    stageB(0, 0);
    stageA(0, 0);
    WAIT_ASYNC();
    __syncthreads();

    int nchunks = K / KC;
    for (int c = 0; c < nchunks; ++c) {
        int cur = c & 1;
        int nxt = cur ^ 1;
        if (c + 1 < nchunks) {                 // prefetch next chunk (async DMA)
            stageB((c + 1) * KC, nxt);
            stageA((c + 1) * KC, nxt);
        }

        if (active) {
            const float* Aw = &As[cur][wave * 16 * SA + lm * SA + kb];
            const float* Bb = &Bs[cur][hi * PS + (lm << 1)];
            for (int k = 0; k < KC; k += 4) {
                v2f a = *(const v2f*)(Aw + k);
                a = a * s;                                   // fold mean scale
                const float* Bk = Bb + (k >> 1) * PS;
#pragma unroll
                for (int t = 0; t < 4; ++t) {
                    v2f b = *(const v2f*)(Bk + (t << 5));
                    acc[t] = __builtin_amdgcn_wmma_f32_16x16x4_f32(
                        false, a, false, b, (short)0, acc[t], false, false);
                }
            }
        }

        WAIT_ASYNC();       // next chunk's DMA done (hidden behind the WMMAs)
        __syncthreads();    // all waves finished reading `cur`, writes visible
    }

    if (active) {
        for (int t = 0; t < 4; ++t)
            for (int i = 0; i < 8; ++i)
                out[(size_t)(m0 + i + 8 * hi) * O + o0 + t * 16 + lm] = acc[t][i];
    }
}

// ---------------------------------------------------------------------------
// log-softmax over rows of 128; one wave per row, shfl_xor reductions
// ---------------------------------------------------------------------------
__global__ void logsoftmax128(float* __restrict__ h, int N) {
    int row = blockIdx.x * (blockDim.x >> 5) + (threadIdx.x >> 5);
    int lane = threadIdx.x & 31;
    if (row >= N) return;
    float* p = h + (size_t)row * 128;
    float v[4];
    float mx = -INFINITY;
    for (int i = 0; i < 4; ++i) {
        v[i] = p[lane + 32 * i];
        mx = fmaxf(mx, v[i]);
    }
    for (int off = 16; off; off >>= 1) mx = fmaxf(mx, __shfl_xor(mx, off, 32));
    float sum = 0.0f;
    for (int i = 0; i < 4; ++i) sum += __expf(v[i] - mx);
    for (int off = 16; off; off >>= 1) sum += __shfl_xor(sum, off, 32);
    float lse = mx + __logf(sum);
    for (int i = 0; i < 4; ++i) p[lane + 32 * i] = v[i] - lse;
}

// ---------------------------------------------------------------------------
extern "C" void kernel_launch(void* const* d_in, const int* in_sizes, int n_in,
                              void* d_out, int out_size, void* d_ws, size_t ws_size,
                              hipStream_t stream) {
    const float* x     = (const float*)d_in[0];   // [N,256]
    const int*   ei    = (const int*)d_in[1];     // [2,E]
    const int*   et    = (const int*)d_in[2];     // [E]
    const float* w1    = (const float*)d_in[3];   // [R,256,512]
    const float* root1 = (const float*)d_in[4];   // [256,512]
    const float* b1    = (const float*)d_in[5];   // [512]
    const float* w2    = (const float*)d_in[6];   // [R,512,128]
    const float* root2 = (const float*)d_in[7];   // [512,128]
    const float* b2    = (const float*)d_in[8];   // [128]

    const int N = NN, E = EE, R = RR;
    float* h   = (float*)d_out;                     // [N,128]
    float* emb = (float*)d_out + (size_t)N * 128;   // [N,512]

    // workspace layout (floats)
    float* inv    = (float*)d_ws;                               // R*N
    float* buf    = inv + (size_t)R * N;                        // N*512
    float* w1p    = buf + (size_t)N * 512;                      // R*256*512
    float* root1p = w1p + (size_t)R * 256 * 512;                // 256*512
    float* w2p    = root1p + (size_t)256 * 512;                 // R*512*128
    float* root2p = w2p + (size_t)R * 512 * 128;                // 512*128

    const int ntiles = N / 16;                      // 3125
    dim3 blk(256);

    // --- pack all weight matrices into paired-row layout ---------------------
    pack_pairs<<<2048, blk, 0, stream>>>(w1, w1p, 256, 512, R);
    pack_pairs<<<512, blk, 0, stream>>>(root1, root1p, 256, 512, 1);
    pack_pairs<<<1024, blk, 0, stream>>>(w2, w2p, 512, 128, R);
    pack_pairs<<<256, blk, 0, stream>>>(root2, root2p, 512, 128, 1);

    // --- per-(rel,dst) edge counts -> reciprocal means -----------------------
    zero_f32<<<2048, blk, 0, stream>>>(inv, R * N);
    count_edges<<<(E + 255) / 256, blk, 0, stream>>>(ei, et, inv, E, N);
    finalize_inv<<<(R * N + 255) / 256, blk, 0, stream>>>(inv, R * N);

    // --- layer 1: x[N,256] -> emb[N,512] ------------------------------------
    {
        const int K = 256, O = 512;
        dim3 grid((ntiles + 7) / 8, O / 64);
        rgcn_gemm_f32<<<grid, blk, 0, stream>>>(x, nullptr, root1p, b1, emb,
                                                ntiles, K, O, 0);
        for (int r = 0; r < R; ++r) {
            zero_f32<<<2048, blk, 0, stream>>>(buf, N * K);
            scatter_rel<<<(E * 32 + 255) / 256, blk, 0, stream>>>(x, ei, et, buf,
                                                                  r, K, E);
            rgcn_gemm_f32<<<grid, blk, 0, stream>>>(buf, inv + (size_t)r * N,
                                                    w1p + (size_t)r * K * O,
                                                    nullptr, emb, ntiles, K, O, 1);
        }
        relu_inplace<<<2048, blk, 0, stream>>>(emb, N * O);
    }

    // --- layer 2: emb[N,512] -> h[N,128] ------------------------------------
    {
        const int K = 512, O = 128;
        dim3 grid((ntiles + 7) / 8, O / 64);
        rgcn_gemm_f32<<<grid, blk, 0, stream>>>(emb, nullptr, root2p, b2, h,
                                                ntiles, K, O, 0);
        for (int r = 0; r < R; ++r) {
            zero_f32<<<2048, blk, 0, stream>>>(buf, N * K);
            scatter_rel<<<(E * 32 + 255) / 256, blk, 0, stream>>>(emb, ei, et, buf,
                                                                  r, K, E);
            rgcn_gemm_f32<<<grid, blk, 0, stream>>>(buf, inv + (size_t)r * N,
                                                    w2p + (size_t)r * K * O,
                                                    nullptr, h, ntiles, K, O, 1);
        }
    }

    // --- log-softmax over rows of 128 ---------------------------------------
    logsoftmax128<<<(N + 7) / 8, blk, 0, stream>>>(h, N);
}